// GNNClassifier_71365176590992
// MI455X (gfx1250) — compile-verified
//
#include <hip/hip_runtime.h>

typedef __attribute__((ext_vector_type(2))) float v2f;
typedef __attribute__((ext_vector_type(8))) float v8f;

#define NC   100000
#define ESIM 600000
#define EAB  300000
#define ETG  300000
#define HD   128

// ---------------------------------------------------------------- utilities
__global__ void zero_kernel(float* __restrict__ p, size_t n) {
  size_t i = (size_t)blockIdx.x * blockDim.x + threadIdx.x;
  size_t stride = (size_t)gridDim.x * blockDim.x;
  for (; i < n; i += stride) p[i] = 0.0f;
}

__global__ void combine3_kernel(const float* __restrict__ a, const float* __restrict__ b,
                                const float* __restrict__ c, float* __restrict__ o, int n) {
  int i = blockIdx.x * blockDim.x + threadIdx.x;
  if (i < n) o[i] = a[i] + b[i] + c[i];
}

__global__ void degree_kernel(const long long* __restrict__ dst, int E, float* __restrict__ deg) {
  int e = blockIdx.x * blockDim.x + threadIdx.x;
  if (e < E) atomicAdd(&deg[(int)dst[e]], 1.0f);
}

// one edge handled by D/4 threads; float4 gather + 4 scalar float atomics
__global__ void scatter_add_kernel(const float* __restrict__ X,
                                   const long long* __restrict__ ei, int E,
                                   int tpe_shift, float* __restrict__ out) {
  const int tpe = 1 << tpe_shift;      // threads per edge
  const int D = tpe << 2;              // feature dim
  const int local = threadIdx.x;
  const int epb = blockDim.x >> tpe_shift;
  const int e = blockIdx.x * epb + (local >> tpe_shift);
  if (e >= E) return;
  const int f = (local & (tpe - 1)) << 2;
  const long long s = ei[e];
  const long long d = ei[E + e];
  const float4 v = *(const float4*)(X + (size_t)s * D + f);
  float* o = out + (size_t)d * D + f;
  atomicAdd(o + 0, v.x);
  atomicAdd(o + 1, v.y);
  atomicAdd(o + 2, v.z);
  atomicAdd(o + 3, v.w);
}

__global__ void divide_rows_kernel(float* __restrict__ sum, const float* __restrict__ deg,
                                   int M, int d_shift) {
  size_t i = (size_t)blockIdx.x * blockDim.x + threadIdx.x;
  size_t n = (size_t)M << d_shift;
  if (i < n) {
    int m = (int)(i >> d_shift);
    sum[i] /= fmaxf(deg[m], 1.0f);
  }
}

// ------------------------------------------------------- fp32 WMMA GEMM, N=128
// C[M x 128] (+)= A[M x K] @ W[K x 128]; optional bias + relu on final pass.
// Block: 256 threads = 8 waves; wave w owns cols [16w,16w+16) for TWO row-tiles
// (m0..m0+15 and m0+16..m0+31) -> two independent WMMA accumulator chains per
// wave (2-way ILP to hide XDL latency) sharing the same B fragments from LDS.
// W staged in dynamic LDS (K*128*4 bytes, <=128KB of the 320KB WGP LDS).
__global__ void wmma_gemm_n128(const float* __restrict__ A, int M, int K,
                               const float* __restrict__ W,
                               float* __restrict__ C,
                               int beta, const float* __restrict__ bias, int relu) {
  extern __shared__ float lds[];
  const int tid = threadIdx.x;
  const int KW = K * 128;
  for (int i = tid; i < KW; i += blockDim.x) lds[i] = W[i];
  __syncthreads();

  const int wave = tid >> 5;
  const int lane = tid & 31;
  const int half = lane >> 4;   // 0: lanes 0-15, 1: lanes 16-31
  const int mr = lane & 15;
  const int m0 = blockIdx.x << 5;   // 32 rows per block
  const int n0 = wave << 4;

  v8f acc0, acc1;
  if (beta) {
#pragma unroll
    for (int r = 0; r < 8; ++r) {
      acc0[r] = C[(size_t)(m0 + half * 8 + r) * 128 + n0 + mr];
      acc1[r] = C[(size_t)(m0 + 16 + half * 8 + r) * 128 + n0 + mr];
    }
  } else {
#pragma unroll
    for (int r = 0; r < 8; ++r) { acc0[r] = 0.0f; acc1[r] = 0.0f; }
  }

  // A 16x4 layout: lanes 0-15 -> K,K+1 ; lanes 16-31 -> K+2,K+3 (row M=mr)
  const float* __restrict__ Arow0 = A + (size_t)(m0 + mr) * K + 2 * half;
  const float* __restrict__ Arow1 = Arow0 + (size_t)16 * K;
  // B 4x16 layout: lanes 0-15 -> rows K,K+1 ; lanes 16-31 -> rows K+2,K+3 (col N=n0+mr)
  const float* __restrict__ Bcol = lds + (size_t)(2 * half) * 128 + n0 + mr;

  for (int k = 0; k < K; k += 4) {
    v2f a0, a1, b;
    a0[0] = Arow0[k];
    a0[1] = Arow0[k + 1];
    a1[0] = Arow1[k];
    a1[1] = Arow1[k + 1];
    b[0] = Bcol[(size_t)k * 128];
    b[1] = Bcol[(size_t)k * 128 + 128];
    acc0 = __builtin_amdgcn_wmma_f32_16x16x4_f32(false, a0, false, b, (short)0, acc0,
                                                 false, false);
    acc1 = __builtin_amdgcn_wmma_f32_16x16x4_f32(false, a1, false, b, (short)0, acc1,
                                                 false, false);
  }

#pragma unroll
  for (int r = 0; r < 8; ++r) {
    float v0 = acc0[r];
    float v1 = acc1[r];
    if (bias) { float bb = bias[n0 + mr]; v0 += bb; v1 += bb; }
    if (relu) { v0 = v0 > 0.0f ? v0 : 0.0f; v1 = v1 > 0.0f ? v1 : 0.0f; }
    C[(size_t)(m0 + half * 8 + r) * 128 + n0 + mr] = v0;
    C[(size_t)(m0 + 16 + half * 8 + r) * 128 + n0 + mr] = v1;
  }
}

// ----------------------------------------------------------------- heads
// One wave per comment row: target/intent logits, sigmoid+softmax, impl head.
__global__ void heads_kernel(const float* __restrict__ h,
                             const float* __restrict__ Wt, const float* __restrict__ bt,
                             const float* __restrict__ Wi, const float* __restrict__ bi,
                             const float* __restrict__ Wm, const float* __restrict__ bm,
                             float* __restrict__ out) {
  const int row = (int)(((size_t)blockIdx.x * blockDim.x + threadIdx.x) >> 5);
  const int lane = threadIdx.x & 31;
  if (row >= NC) return;

  const float4 h4 = *(const float4*)(h + (size_t)row * HD + lane * 4);
  const float hv[4] = {h4.x, h4.y, h4.z, h4.w};
  const int r0 = lane * 4;

  float tl[5];
#pragma unroll
  for (int t = 0; t < 5; ++t) {
    float p = 0.0f;
#pragma unroll
    for (int j = 0; j < 4; ++j) p += hv[j] * Wt[(r0 + j) * 5 + t];
#pragma unroll
    for (int o = 16; o > 0; o >>= 1) p += __shfl_xor(p, o, 32);
    tl[t] = p + bt[t];
  }
  float il[4];
#pragma unroll
  for (int t = 0; t < 4; ++t) {
    float p = 0.0f;
#pragma unroll
    for (int j = 0; j < 4; ++j) p += hv[j] * Wi[(r0 + j) * 4 + t];
#pragma unroll
    for (int o = 16; o > 0; o >>= 1) p += __shfl_xor(p, o, 32);
    il[t] = p + bi[t];
  }

  float tp[5];
#pragma unroll
  for (int t = 0; t < 5; ++t) tp[t] = 1.0f / (1.0f + __expf(-tl[t]));

  float mx = il[0];
#pragma unroll
  for (int t = 1; t < 4; ++t) mx = fmaxf(mx, il[t]);
  float ip[4], den = 0.0f;
#pragma unroll
  for (int t = 0; t < 4; ++t) { ip[t] = __expf(il[t] - mx); den += ip[t]; }
#pragma unroll
  for (int t = 0; t < 4; ++t) ip[t] /= den;

  float ml[3];
#pragma unroll
  for (int m = 0; m < 3; ++m) {
    float p = 0.0f;
#pragma unroll
    for (int j = 0; j < 4; ++j) p += hv[j] * Wm[(r0 + j) * 3 + m];
#pragma unroll
    for (int o = 16; o > 0; o >>= 1) p += __shfl_xor(p, o, 32);
    float extra = bm[m];
#pragma unroll
    for (int t = 0; t < 5; ++t) extra += tp[t] * Wm[(128 + t) * 3 + m];
#pragma unroll
    for (int t = 0; t < 4; ++t) extra += ip[t] * Wm[(133 + t) * 3 + m];
    ml[m] = p + extra;
  }

  if (lane == 0) {
    float* ot = out + (size_t)row * 5;
#pragma unroll
    for (int t = 0; t < 5; ++t) ot[t] = tl[t];
    float* oi = out + (size_t)NC * 5 + (size_t)row * 4;
#pragma unroll
    for (int t = 0; t < 4; ++t) oi[t] = il[t];
    float* om = out + (size_t)NC * 9 + (size_t)row * 3;
#pragma unroll
    for (int m = 0; m < 3; ++m) om[m] = ml[m];
  }
}

// ----------------------------------------------------------------- launch
extern "C" void kernel_launch(void* const* d_in, const int* in_sizes, int n_in,
                              void* d_out, int out_size, void* d_ws, size_t ws_size,
                              hipStream_t stream) {
  (void)in_sizes; (void)n_in; (void)out_size; (void)ws_size;

  const float* x_comment = (const float*)d_in[0];
  const float* x_topic   = (const float*)d_in[1];
  const float* x_claim   = (const float*)d_in[2];
  const float* c1s_Wl = (const float*)d_in[3];
  const float* c1s_bl = (const float*)d_in[4];
  const float* c1s_Wr = (const float*)d_in[5];
  const float* c1a_Wl = (const float*)d_in[6];
  const float* c1a_bl = (const float*)d_in[7];
  const float* c1a_Wr = (const float*)d_in[8];
  const float* c1t_Wl = (const float*)d_in[9];
  const float* c1t_bl = (const float*)d_in[10];
  const float* c1t_Wr = (const float*)d_in[11];
  const float* c2s_Wl = (const float*)d_in[12];
  const float* c2s_bl = (const float*)d_in[13];
  const float* c2s_Wr = (const float*)d_in[14];
  const float* c2a_Wl = (const float*)d_in[15];
  const float* c2a_bl = (const float*)d_in[16];
  const float* c2a_Wr = (const float*)d_in[17];
  const float* c2t_Wl = (const float*)d_in[18];
  const float* c2t_bl = (const float*)d_in[19];
  const float* c2t_Wr = (const float*)d_in[20];
  const float* Wt = (const float*)d_in[21];
  const float* bt = (const float*)d_in[22];
  const float* Wi = (const float*)d_in[23];
  const float* bi = (const float*)d_in[24];
  const float* Wm = (const float*)d_in[25];
  const float* bm = (const float*)d_in[26];
  const long long* ei_sim     = (const long long*)d_in[27];
  const long long* ei_about   = (const long long*)d_in[28];
  const long long* ei_targets = (const long long*)d_in[29];
  float* out = (float*)d_out;

  // workspace layout (floats)
  float* ws = (float*)d_ws;
  float* mean_c  = ws;                                   // NC*256
  float* mean_ab = mean_c  + (size_t)NC * 256;           // NC*128
  float* mean_tg = mean_ab + (size_t)NC * 128;           // NC*128
  float* h1      = mean_tg + (size_t)NC * 128;           // NC*128
  float* deg_sim = h1      + (size_t)NC * 128;           // NC
  float* deg_ab  = deg_sim + NC;                         // NC
  float* deg_tg  = deg_ab  + NC;                         // NC
  float* wr1     = deg_tg  + NC;                         // 256*128
  float* wr2     = wr1 + 256 * 128;                      // 128*128
  float* bc1     = wr2 + 128 * 128;                      // 128
  float* bc2     = bc1 + 128;                            // 128
  // reuse of the big mean_c region after layer 1:
  float* mean_sim2 = mean_c;                             // NC*128
  float* h2        = mean_c + (size_t)NC * 128;          // NC*128

  // zero accumulators + degrees
  zero_kernel<<<4096, 256, 0, stream>>>(mean_c, (size_t)NC * 512);
  zero_kernel<<<256, 256, 0, stream>>>(deg_sim, (size_t)NC * 3);

  // fold the three Wr / bl per layer into one
  combine3_kernel<<<(256 * 128 + 255) / 256, 256, 0, stream>>>(c1s_Wr, c1a_Wr, c1t_Wr, wr1, 256 * 128);
  combine3_kernel<<<(128 * 128 + 255) / 256, 256, 0, stream>>>(c2s_Wr, c2a_Wr, c2t_Wr, wr2, 128 * 128);
  combine3_kernel<<<1, 128, 0, stream>>>(c1s_bl, c1a_bl, c1t_bl, bc1, 128);
  combine3_kernel<<<1, 128, 0, stream>>>(c2s_bl, c2a_bl, c2t_bl, bc2, 128);

  // degrees
  degree_kernel<<<(ESIM + 255) / 256, 256, 0, stream>>>(ei_sim + ESIM, ESIM, deg_sim);
  degree_kernel<<<(EAB + 255) / 256, 256, 0, stream>>>(ei_about + EAB, EAB, deg_ab);
  degree_kernel<<<(ETG + 255) / 256, 256, 0, stream>>>(ei_targets + ETG, ETG, deg_tg);

  // layer-1 segment sums -> means (topic/claim means reused by layer 2)
  scatter_add_kernel<<<(ESIM + 3) / 4, 256, 0, stream>>>(x_comment, ei_sim, ESIM, 6, mean_c);
  scatter_add_kernel<<<(EAB + 7) / 8, 256, 0, stream>>>(x_topic, ei_about, EAB, 5, mean_ab);
  scatter_add_kernel<<<(ETG + 7) / 8, 256, 0, stream>>>(x_claim, ei_targets, ETG, 5, mean_tg);
  divide_rows_kernel<<<(NC * 256 + 255) / 256, 256, 0, stream>>>(mean_c, deg_sim, NC, 8);
  divide_rows_kernel<<<(NC * 128 + 255) / 256, 256, 0, stream>>>(mean_ab, deg_ab, NC, 7);
  divide_rows_kernel<<<(NC * 128 + 255) / 256, 256, 0, stream>>>(mean_tg, deg_tg, NC, 7);

  // layer 1: h1 = relu(mean_sim@Wl_s + x_c@(sum Wr) + mean_ab@Wl_a + mean_tg@Wl_t + sum bl)
  const int mg = NC / 32;  // 3125, exact (32 rows per block)
  wmma_gemm_n128<<<mg, 256, 256 * 128 * 4, stream>>>(mean_c,    NC, 256, c1s_Wl, h1, 0, nullptr, 0);
  wmma_gemm_n128<<<mg, 256, 256 * 128 * 4, stream>>>(x_comment, NC, 256, wr1,    h1, 1, nullptr, 0);
  wmma_gemm_n128<<<mg, 256, 128 * 128 * 4, stream>>>(mean_ab,   NC, 128, c1a_Wl, h1, 1, nullptr, 0);
  wmma_gemm_n128<<<mg, 256, 128 * 128 * 4, stream>>>(mean_tg,   NC, 128, c1t_Wl, h1, 1, bc1, 1);

  // layer 2: sim mean of h1, then GEMMs into h2
  zero_kernel<<<4096, 256, 0, stream>>>(mean_sim2, (size_t)NC * 128);
  scatter_add_kernel<<<(ESIM + 7) / 8, 256, 0, stream>>>(h1, ei_sim, ESIM, 5, mean_sim2);
  divide_rows_kernel<<<(NC * 128 + 255) / 256, 256, 0, stream>>>(mean_sim2, deg_sim, NC, 7);

  wmma_gemm_n128<<<mg, 256, 128 * 128 * 4, stream>>>(mean_sim2, NC, 128, c2s_Wl, h2, 0, nullptr, 0);
  wmma_gemm_n128<<<mg, 256, 128 * 128 * 4, stream>>>(h1,        NC, 128, wr2,    h2, 1, nullptr, 0);
  wmma_gemm_n128<<<mg, 256, 128 * 128 * 4, stream>>>(mean_ab,   NC, 128, c2a_Wl, h2, 1, nullptr, 0);
  wmma_gemm_n128<<<mg, 256, 128 * 128 * 4, stream>>>(mean_tg,   NC, 128, c2t_Wl, h2, 1, bc2, 1);

  // fused heads -> d_out (target 5 | intent 4 | impl 3, concatenated flat)
  heads_kernel<<<(NC * 32 + 255) / 256, 256, 0, stream>>>(h2, Wt, bt, Wi, bi, Wm, bm, out);
}